// AttentionModel_32530082300297
// MI455X (gfx1250) — compile-verified
//
#include <hip/hip_runtime.h>
#include <hip/hip_bf16.h>

// Fused pointer-network decode step for MI455X (gfx1250).
// One block per batch row b (B=256 blocks, 256 threads = 8 wave32).
// Heavy GEMMs (emb @ W_node) run on v_wmma_f32_16x16x4_f32 (native fp32 WMMA).
// K/V/logit_K never touch HBM: embeddings (131MB) stay L2-resident, logits in LDS.
// Embedding tiles are double-buffered via GLOBAL_LOAD_ASYNC_TO_LDS_B128 (ASYNCcnt),
// hiding L2 latency behind a full tile of WMMA + online-softmax work.

typedef __attribute__((ext_vector_type(2))) float v2f;
typedef __attribute__((ext_vector_type(8))) float v8f;

#define DD 128
#define EPAD 132      // emb tile row pitch (floats): 528B rows, 16B-aligned chunks
#define WPAD 130      // W column pitch (floats), conflict-free ds_load_b64
#define NEGBIG (-1.0e9f)

#define ASYNC_WAIT2() asm volatile("s_wait_asynccnt 0x2" ::: "memory")
#define ASYNC_WAIT0() asm volatile("s_wait_asynccnt 0x0" ::: "memory")

__device__ __forceinline__ float groupSum16(float v) {
    v += __shfl_xor(v, 1);
    v += __shfl_xor(v, 2);
    v += __shfl_xor(v, 4);
    v += __shfl_xor(v, 8);
    return v;   // sum within each 16-lane half of the wave32
}

// Stage one 16x128 f32 tile (rows n0..n0+15, clamped to <N) into LDS via the
// async-to-LDS engine: 512 x 16B chunks, 2 per thread -> 2 ASYNCcnt per wave.
__device__ __forceinline__ void asyncStageTile(const float* __restrict__ eb,
                                               int n0, int N,
                                               float* dstLds, int t) {
#pragma unroll
    for (int j = 0; j < 2; ++j) {
        int i = t + j * 256;            // 16B chunk id, 0..511
        int r = i >> 5;                 // tile row 0..15
        int c = (i & 31) << 2;          // float column 0..124
        int row = n0 + r;
        if (row > N - 1) row = N - 1;   // clamp: keep address legal; tail masked later
        unsigned long long ga = (unsigned long long)(uintptr_t)(eb + (size_t)row * DD + c);
        unsigned int       la = (unsigned int)(uintptr_t)(dstLds + r * EPAD + c);
        asm volatile("global_load_async_to_lds_b128 %0, %1, off"
                     :: "v"(la), "v"(ga) : "memory");
    }
}

__global__ __launch_bounds__(256)
void ptrnet_decode_kernel(const float* __restrict__ emb,   // [B,N,128]
                          const float* __restrict__ Wn,    // [128,384]
                          const float* __restrict__ Wf,    // [128,128]
                          const float* __restrict__ Ws,    // [256,128]
                          const float* __restrict__ Wo,    // [128,128]
                          const int*   __restrict__ firstIdx,
                          const int*   __restrict__ lastIdx,
                          const unsigned char* __restrict__ maskB, // [B,N] bool
                          float* __restrict__ out,         // [B,N]
                          int N) {
    __shared__ float wLds[256 * WPAD];        // 133 KB: W_node cols, column-major
    __shared__ float embLds[2][16 * EPAD];    // 16.9 KB: double-buffered emb tiles
    __shared__ float logitsLds[1008];
    __shared__ float redLds[256];             // phase A partial sums + phase F reduce
    __shared__ float geLds[DD];
    __shared__ float ctxLds[2 * DD];
    __shared__ float qLds[DD];
    __shared__ float headsLds[DD];
    __shared__ float glimpseLds[DD];
    __shared__ float pLds[128];
    __shared__ int   maskLds[2][16];

    const int b    = blockIdx.x;
    const int t    = threadIdx.x;
    const int lane = t & 31;
    const int wv   = t >> 5;        // wave id == head id (0..7)
    const int l16  = lane & 15;
    const int hi   = lane >> 4;     // which 16-lane half
    const int akk  = hi << 1;       // K-offset of this half in A/B fragments
    const int NT   = (N + 15) >> 4;

    const float* eb = emb + (size_t)b * N * DD;

    // ---------- Phase A: graph_embed = mean_n emb[b], gather first/last ----------
    {
        int dcol = t & (DD - 1);
        int half = t >> 7;
        float acc = 0.f;
        for (int n = half; n < N; n += 2) acc += eb[(size_t)n * DD + dcol];
        redLds[t] = acc;
    }
    const int fi = firstIdx[b], li = lastIdx[b];
    __syncthreads();
    if (t < DD) {
        geLds[t]  = (redLds[t] + redLds[t + DD]) * (1.0f / (float)N);
        ctxLds[t] = eb[(size_t)fi * DD + t];
    } else {
        ctxLds[t] = eb[(size_t)li * DD + (t - DD)];
    }
    __syncthreads();

    // ---------- Phase B: q = ge @ W_fixed + [e_first|e_last] @ W_step ----------
    if (t < DD) {
        float f = 0.f;
        for (int k = 0; k < DD; ++k) f += geLds[k] * Wf[k * DD + t];
        float s = 0.f;
        for (int k = 0; k < 2 * DD; ++k) s += ctxLds[k] * Ws[k * DD + t];
        qLds[t] = f + s;
    }
    // stage W_node cols 0..255 (K|V) column-major into LDS (coalesced over cols)
    for (int k = 0; k < DD; ++k)
        wLds[t * WPAD + k] = Wn[k * 384 + t];

    // async-prefetch emb tile 0 + mask 0 (consumed after the barrier below)
    asyncStageTile(eb, 0, N, embLds[0], t);
    if (t < 16) maskLds[0][t] = (t < N) ? (int)maskB[(size_t)b * N + t] : 1;
    __syncthreads();

    // ---------- Phase C: fused K/V WMMA + online-softmax multi-head glimpse ----------
    // Wave wv owns head wv: K-tile cols [16wv,16wv+16), V-tile cols [128+16wv, ...).
    const float qv = qLds[wv * 16 + l16];
    float mRun = -3.0e38f, sRun = 0.f, hAcc = 0.f;   // per-half online state
    const float* bKBase = &wLds[(wv * 16 + l16) * WPAD];
    const float* bVBase = &wLds[(128 + wv * 16 + l16) * WPAD];

    for (int tile = 0; tile < NT; ++tile) {
        const int n0  = tile << 4;
        const int cur = tile & 1;
        __syncthreads();   // all readers of buf[cur^1] (tile-1) are done
        // stage tile+1 one full iteration ahead
        asyncStageTile(eb, n0 + 16, N, embLds[cur ^ 1], t);
        if (t < 16) {
            int nn = n0 + 16 + t;
            maskLds[cur ^ 1][t] = (nn < N) ? (int)maskB[(size_t)b * N + nn] : 1;
        }
        ASYNC_WAIT2();     // tile's own 2 async chunks have landed (in-order cnt)
        __syncthreads();   // every wave's chunks for this tile now visible

        const float* aBase = &embLds[cur][l16 * EPAD];
        v8f accK = {}; v8f accV = {};
#pragma unroll
        for (int k0 = 0; k0 < DD; k0 += 4) {
            v2f a  = *(const v2f*)(aBase  + k0 + akk);
            v2f bk = *(const v2f*)(bKBase + k0 + akk);
            v2f bv = *(const v2f*)(bVBase + k0 + akk);
            accK = __builtin_amdgcn_wmma_f32_16x16x4_f32(false, a, false, bk, (short)0, accK, false, false);
            accV = __builtin_amdgcn_wmma_f32_16x16x4_f32(false, a, false, bv, (short)0, accV, false, false);
        }

        // compat + online softmax: C-layout row r holds n = r (low half) / r+8 (high),
        // lane holds dk = l16. Each half keeps its own (m, s, hAcc).
#pragma unroll
        for (int r = 0; r < 8; ++r) {
            float c = groupSum16(accK[r] * qv) * 0.25f;      // 1/sqrt(dk=16)
            int nl = r + (hi << 3);
            if (n0 + nl >= N || maskLds[cur][nl]) c = NEGBIG;
            float mNew = fmaxf(mRun, c);
            float al = __expf(mRun - mNew);
            float p  = __expf(c - mNew);
            sRun = sRun * al + p;
            hAcc = hAcc * al + p * accV[r];
            mRun = mNew;
        }
    }
    ASYNC_WAIT0();         // drain speculative issue before embLds reuse
    {   // merge the two half-wave softmax states
        float mO = __shfl_xor(mRun, 16);
        float sO = __shfl_xor(sRun, 16);
        float hO = __shfl_xor(hAcc, 16);
        float mS = fmaxf(mRun, mO);
        float e1 = __expf(mRun - mS), e2 = __expf(mO - mS);
        float S  = sRun * e1 + sO * e2;
        float hv = (hAcc * e1 + hO * e2) / S;
        if (lane < 16) headsLds[wv * 16 + lane] = hv;
    }
    __syncthreads();       // phase C fully done (embLds free, headsLds published)

    // ---------- Phase D: glimpse = heads @ W_out; reload logit_K slice;
    //            meanwhile async-prefetch phase-E tile 0 ----------
    asyncStageTile(eb, 0, N, embLds[0], t);
    if (t < 16) maskLds[0][t] = (t < N) ? (int)maskB[(size_t)b * N + t] : 1;
    if (t < DD) {
        float g = 0.f;
        for (int k = 0; k < DD; ++k) g += headsLds[k] * Wo[k * DD + t];
        glimpseLds[t] = g;
    }
    {   // W_node cols 256..383 -> wLds cols 0..127 (all 256 threads, split k)
        int c = t & 127, kh = t >> 7;
        for (int k = kh * 64; k < (kh + 1) * 64; ++k)
            wLds[c * WPAD + k] = Wn[k * 384 + 256 + c];
    }
    __syncthreads();

    // ---------- Phase E: logit_K WMMA + tanh-clipped masked logits ----------
    const float gv = glimpseLds[wv * 16 + l16];
    const float* bLBase = &wLds[(wv * 16 + l16) * WPAD];

    for (int tile = 0; tile < NT; ++tile) {
        const int n0  = tile << 4;
        const int cur = tile & 1;
        __syncthreads();
        asyncStageTile(eb, n0 + 16, N, embLds[cur ^ 1], t);
        if (t < 16) {
            int nn = n0 + 16 + t;
            maskLds[cur ^ 1][t] = (nn < N) ? (int)maskB[(size_t)b * N + nn] : 1;
        }
        ASYNC_WAIT2();
        __syncthreads();

        const float* aBase = &embLds[cur][l16 * EPAD];
        v8f accL = {};
#pragma unroll
        for (int k0 = 0; k0 < DD; k0 += 4) {
            v2f a  = *(const v2f*)(aBase  + k0 + akk);
            v2f bl = *(const v2f*)(bLBase + k0 + akk);
            accL = __builtin_amdgcn_wmma_f32_16x16x4_f32(false, a, false, bl, (short)0, accL, false, false);
        }
#pragma unroll
        for (int r = 0; r < 8; ++r) {
            float v = groupSum16(accL[r] * gv);    // partial over this wave's 16 d's
            if (l16 == 0) pLds[wv * 16 + r + (hi << 3)] = v;
        }
        __syncthreads();
        if (t < 16) {
            float l = 0.f;
            for (int w = 0; w < 8; ++w) l += pLds[w * 16 + t];
            l = tanhf(l * 0.08838834764831843f) * 10.0f;   // /sqrt(128), tanh clip
            int ng = n0 + t;
            if (ng < N) logitsLds[ng] = maskLds[cur][t] ? NEGBIG : l;
        }
    }
    ASYNC_WAIT0();         // drain before s_endpgm-era LDS reuse below
    __syncthreads();

    // ---------- Phase F: log_softmax over N, write output ----------
    float lm = -3.0e38f;
    for (int i = t; i < N; i += 256) lm = fmaxf(lm, logitsLds[i]);
    redLds[t] = lm;
    __syncthreads();
    for (int s2 = 128; s2 > 0; s2 >>= 1) {
        if (t < s2) redLds[t] = fmaxf(redLds[t], redLds[t + s2]);
        __syncthreads();
    }
    const float M = redLds[0];
    __syncthreads();
    float ls = 0.f;
    for (int i = t; i < N; i += 256) ls += __expf(logitsLds[i] - M);
    redLds[t] = ls;
    __syncthreads();
    for (int s2 = 128; s2 > 0; s2 >>= 1) {
        if (t < s2) redLds[t] += redLds[t + s2];
        __syncthreads();
    }
    const float lse = M + __logf(redLds[0]);
    float* ob = out + (size_t)b * N;
    for (int i = t; i < N; i += 256) ob[i] = logitsLds[i] - lse;
}

extern "C" void kernel_launch(void* const* d_in, const int* in_sizes, int n_in,
                              void* d_out, int out_size, void* d_ws, size_t ws_size,
                              hipStream_t stream) {
    const float* emb = (const float*)d_in[0];
    const float* Wn  = (const float*)d_in[1];
    const float* Wf  = (const float*)d_in[2];
    const float* Ws  = (const float*)d_in[3];
    const float* Wo  = (const float*)d_in[4];
    const int*   fi  = (const int*)d_in[5];
    const int*   li  = (const int*)d_in[6];
    const unsigned char* mk = (const unsigned char*)d_in[7];  // jnp.bool_ -> 1 byte

    const int B = in_sizes[5];          // first_idx has B elements
    const int N = in_sizes[7] / B;      // mask is [B, N]

    ptrnet_decode_kernel<<<dim3(B), dim3(256), 0, stream>>>(
        emb, Wn, Wf, Ws, Wo, fi, li, mk, (float*)d_out, N);

    (void)n_in; (void)out_size; (void)d_ws; (void)ws_size;
}